// MbpGINEMessagePassing_53833120088740
// MI455X (gfx1250) — compile-verified
//
#include <hip/hip_runtime.h>

typedef __attribute__((ext_vector_type(2))) float v2f;
typedef __attribute__((ext_vector_type(8))) float v8f;

// ---------------------------------------------------------------------------
// WMMA fp32 GEMM:  out[M x N] = A[M x K=128] @ W[N x K=128]^T + bias[N]
//                  (+ add[M x N] when ADD, compile-time)
// One 16x16 output tile per wave; K-loop of 32 x v_wmma_f32_16x16x4_f32.
// Fragment layouts per CDNA5 ISA 7.12.2:
//   A 16x4 : lane l -> m = l&15, holds A[m][2*(l>>4)] , A[m][2*(l>>4)+1]
//   B 4x16 : lane l -> n = l&15, holds B[2*(l>>4)][n] , B[2*(l>>4)+1][n]
//            (B = W^T, so per-lane this is W[n][2h], W[n][2h+1] -> contiguous)
//   C/D    : vgpr i -> row i + 8*(l>>4), col l&15
// ---------------------------------------------------------------------------
template <bool ADD>
__global__ __launch_bounds__(256) void gemm_xwT_wmma(
    const float* __restrict__ A, const float* __restrict__ W,
    const float* __restrict__ bias, const float* __restrict__ add,
    float* __restrict__ out, int M, int N) {
  const int K = 128;
  const int lane = threadIdx.x & 31;
  const int wave = threadIdx.x >> 5;
  const int wavesPerBlock = blockDim.x >> 5;
  const int nTiles = N >> 4;
  const long job = (long)blockIdx.x * wavesPerBlock + wave;
  const int tileM = (int)(job / nTiles);
  const int tileN = (int)(job % nTiles);
  if (tileM * 16 >= M) return;

  const int idx16 = lane & 15;
  const int khalf = (lane >> 4) * 2;       // 0 or 2
  const int mrow = tileM * 16 + idx16;     // A row for this lane
  const int ncol = tileN * 16 + idx16;     // W row / output col for this lane

  const float* ap = A + (size_t)mrow * K + khalf;
  const float* bp = W + (size_t)ncol * K + khalf;

  const float binit = bias[ncol];
  v8f c;
#pragma unroll
  for (int i = 0; i < 8; ++i) c[i] = binit;

#pragma unroll 8
  for (int k = 0; k < K; k += 4) {
    v2f a = *(const v2f*)(ap + k);
    v2f b = *(const v2f*)(bp + k);
    c = __builtin_amdgcn_wmma_f32_16x16x4_f32(false, a, false, b,
                                              (short)0, c, false, false);
  }

  const int row0 = tileM * 16 + 8 * (lane >> 4);
#pragma unroll
  for (int i = 0; i < 8; ++i) {
    size_t o = (size_t)(row0 + i) * N + ncol;
    if (ADD) out[o] = c[i] + add[o];
    else     out[o] = c[i];
  }
}

// ---------------------------------------------------------------------------
// Fused edge kernel: per block = 16 edges.
//  phase1: async-DMA conn_feat tile (16x128) into LDS via
//          GLOBAL_LOAD_ASYNC_TO_LDS_B128 (ASYNCcnt), load dst/src indices
//  phase2: Eh tile (16x256) = tile @ We^T + Eb via WMMA -> LDS (Ew | Eb)
//  phase3: conn = relu(signed_sqrt((Qh[dst]+Kh[src])*Ew) + Eb); write conn,
//          atomic segment-sums of conn and Vh[src] into node accumulators.
// ---------------------------------------------------------------------------
__global__ __launch_bounds__(256) void edge_fused_wmma(
    const float* __restrict__ conn_feat, const float* __restrict__ We,
    const float* __restrict__ Ebias, const float* __restrict__ qkv,
    const int* __restrict__ edge_index, float* __restrict__ conn_out,
    float* __restrict__ agg, float* __restrict__ esum,
    int n_edges, int n_nodes) {
  const int K = 128;
  const int LDSW = 132;                     // padded row stride (floats)
  __shared__ float sA[16 * LDSW];           // conn_feat tile
  __shared__ float sEw[16 * LDSW];          // Eh cols 0..127
  __shared__ float sEb[16 * LDSW];          // Eh cols 128..255
  __shared__ int sIdx[32];                  // dst[0..15], src[16..31]

  const int tid = threadIdx.x;
  const int e0 = blockIdx.x * 16;

  // phase 1: async global->LDS DMA of the 16x128 tile (2 x b128 per thread)
#pragma unroll
  for (int it = tid; it < 512; it += 256) {
    int r = it >> 5;                        // 32 float4 per 128-col row
    int cc = (it & 31) * 4;
    const float* gp = conn_feat + (size_t)(e0 + r) * K + cc;
    unsigned lp = (unsigned)(uintptr_t)(&sA[r * LDSW + cc]); // LDS byte offset
    asm volatile("global_load_async_to_lds_b128 %0, %1, off"
                 :: "v"(lp), "v"(gp) : "memory");
  }
  if (tid < 16) sIdx[tid] = edge_index[e0 + tid];                             // dst
  else if (tid < 32) sIdx[tid] = edge_index[(size_t)n_edges + e0 + tid - 16]; // src
  asm volatile("s_wait_asynccnt 0" ::: "memory");  // drain this wave's DMA
  __syncthreads();

  // phase 2: 8 waves x 2 N-tiles each -> 16 tiles covering n = 0..255
  const int lane = tid & 31;
  const int wave = tid >> 5;
  const int idx16 = lane & 15;
  const int khalf = (lane >> 4) * 2;
  const float* apl = &sA[idx16 * LDSW + khalf];

#pragma unroll
  for (int tt = 0; tt < 2; ++tt) {
    const int t = wave * 2 + tt;            // 0..15
    const int ng = t * 16 + idx16;          // global n in 0..255
    const float* bp = We + (size_t)ng * K + khalf;
    const float binit = Ebias[ng];
    v8f c;
#pragma unroll
    for (int i = 0; i < 8; ++i) c[i] = binit;
#pragma unroll 8
    for (int k = 0; k < K; k += 4) {
      v2f a = *(const v2f*)(apl + k);
      v2f b = *(const v2f*)(bp + k);
      c = __builtin_amdgcn_wmma_f32_16x16x4_f32(false, a, false, b,
                                                (short)0, c, false, false);
    }
    float* dl = (t < 8) ? sEw : sEb;
    const int cbase = (t & 7) * 16 + idx16;
    const int r0 = 8 * (lane >> 4);
#pragma unroll
    for (int i = 0; i < 8; ++i) dl[(r0 + i) * LDSW + cbase] = c[i];
  }
  __syncthreads();

  // phase 3: 16 edges x 128 cols; 2 edges per pass, 8 passes
  float* conn_tile = conn_out + (size_t)e0 * 128;
#pragma unroll
  for (int pass = 0; pass < 8; ++pass) {
    const int e = pass * 2 + (tid >> 7);
    const int cc = tid & 127;
    const int d = sIdx[e];
    const int s = sIdx[16 + e];
    const float q = qkv[(size_t)d * 384 + cc];          // Qh[dst]
    const float kk = qkv[(size_t)s * 384 + 128 + cc];   // Kh[src]
    const float v = qkv[(size_t)s * 384 + 256 + cc];    // Vh[src]
    const float ew = sEw[e * LDSW + cc];
    const float eb = sEb[e * LDSW + cc];
    const float c1 = (q + kk) * ew;
    const float ss = copysignf(sqrtf(fabsf(c1)), c1);   // signed sqrt
    const float cn = fmaxf(ss + eb, 0.0f);              // relu(+Eb)
    conn_tile[e * 128 + cc] = cn;
    atomicAdd(&esum[(size_t)d * 128 + cc], cn);
    atomicAdd(&agg[(size_t)d * 128 + cc], v);
  }
}

// ---------------------------------------------------------------------------
extern "C" void kernel_launch(void* const* d_in, const int* in_sizes, int n_in,
                              void* d_out, int out_size, void* d_ws, size_t ws_size,
                              hipStream_t stream) {
  const float* x         = (const float*)d_in[0];
  const float* conn_feat = (const float*)d_in[1];
  const float* qkv_w     = (const float*)d_in[2];
  const float* qkv_b     = (const float*)d_in[3];
  const float* E_w       = (const float*)d_in[4];
  const float* E_b       = (const float*)d_in[5];
  const float* cl_w      = (const float*)d_in[6];
  const float* cl_b      = (const float*)d_in[7];
  const int*   eidx      = (const int*)d_in[8];

  const int n_nodes = in_sizes[0] / 128;   // 50000
  const int n_edges = in_sizes[1] / 128;   // 800000

  // workspace: qkv (N x 384) | agg (N x 128) | esum (N x 128)
  float* qkv  = (float*)d_ws;
  float* agg  = qkv + (size_t)n_nodes * 384;
  float* esum = agg + (size_t)n_nodes * 128;

  float* No   = (float*)d_out;                       // N x 128
  float* conn = No + (size_t)n_nodes * 128;          // E x 128

  // zero the atomic accumulators (every call; harness poisons ws once)
  hipMemsetAsync(agg, 0, (size_t)n_nodes * 256 * sizeof(float), stream);

  const int wavesPerBlock = 8;             // 256 threads
  // 1) QKV GEMM: M = n_nodes, N = 384
  {
    long jobs = (long)(n_nodes / 16) * (384 / 16);
    int blocks = (int)((jobs + wavesPerBlock - 1) / wavesPerBlock);
    gemm_xwT_wmma<false><<<blocks, 256, 0, stream>>>(x, qkv_w, qkv_b, nullptr,
                                                     qkv, n_nodes, 384);
  }
  // 2) fused edge kernel: Eh GEMM + edge math + segment sums
  {
    int blocks = n_edges / 16;
    edge_fused_wmma<<<blocks, 256, 0, stream>>>(conn_feat, E_w, E_b, qkv, eidx,
                                                conn, agg, esum,
                                                n_edges, n_nodes);
  }
  // 3) No = agg + esum @ Wc^T + bc : M = n_nodes, N = 128
  {
    long jobs = (long)(n_nodes / 16) * (128 / 16);
    int blocks = (int)((jobs + wavesPerBlock - 1) / wavesPerBlock);
    gemm_xwT_wmma<true><<<blocks, 256, 0, stream>>>(esum, cl_w, cl_b, agg,
                                                    No, n_nodes, 128);
  }
}